// auxiliary_y_fixed_24103356465521
// MI455X (gfx1250) — compile-verified
//
#include <hip/hip_runtime.h>

typedef __attribute__((ext_vector_type(2))) float v2f;
typedef __attribute__((ext_vector_type(8))) float v8f;

// Monotone float -> uint key: order-preserving, invertible.
__device__ __forceinline__ unsigned fkey(float x) {
  unsigned b = __float_as_uint(x);
  return (b & 0x80000000u) ? ~b : (b | 0x80000000u);
}
__device__ __forceinline__ float fkeyinv(unsigned k) {
  return (k & 0x80000000u) ? __uint_as_float(k & 0x7fffffffu)
                           : __uint_as_float(~k);
}

__global__ void seg_init_kernel(unsigned long long* __restrict__ seg, int B) {
  int i = blockIdx.x * blockDim.x + threadIdx.x;
  if (i < B) seg[i] = ~0ULL;
}

// One wave per 16 rows. GEMV via V_WMMA_F32_16X16X4_F32 with W replicated
// across all 16 B-columns; C[m, n] is identical for all n and equals
// dot(z[base+m], W). Lane 0 holds rows 0..7 in its 8 C VGPRs, lane 16 holds
// rows 8..15 (per the documented 16x16 f32 C/D layout).
__global__ void gemv_seg_kernel(const float* __restrict__ z,
                                const int* __restrict__ bag,
                                const float* __restrict__ W,
                                const float* __restrict__ bptr,
                                float* __restrict__ loc_out,
                                unsigned long long* __restrict__ seg,
                                int* __restrict__ sstart,
                                int* __restrict__ multi,
                                int N, int D) {
  int tid  = blockIdx.x * blockDim.x + threadIdx.x;
  int wid  = tid >> 5;
  int lane = threadIdx.x & 31;
  int base = wid * 16;
  if (base >= N) return;          // wave-uniform: EXEC stays all-1s for WMMA

  int half = lane >> 4;           // 0: K={0,1}, 1: K={2,3} within each chunk
  const float* arow = z + (size_t)(base + (lane & 15)) * D + half * 2;
  const float* wrow = W + half * 2;

  v8f c = {};
#pragma unroll 8
  for (int k0 = 0; k0 < D; k0 += 4) {
    v2f a  = *(const v2f*)(arow + k0);   // A tile 16x4, f32 layout
    v2f bb = *(const v2f*)(wrow + k0);   // B tile 4x16, all columns = W chunk
    c = __builtin_amdgcn_wmma_f32_16x16x4_f32(
        /*neg_a=*/false, a, /*neg_b=*/false, bb,
        /*c_mod=*/(short)0, c, /*reuse_a=*/false, /*reuse_b=*/false);
  }

  float bias = bptr[0];
  if ((lane & 15) == 0) {         // lane 0 -> rows 0..7, lane 16 -> rows 8..15
    int row0 = base + half * 8;
#pragma unroll
    for (int r = 0; r < 8; ++r) {
      int i = row0 + r;
      float v = c[r] + bias;
      loc_out[i] = v;
      int bg = bag[i];
      // max-v with min-index tiebreak as a single 64-bit atomicMin
      unsigned long long pk =
          ((unsigned long long)(~fkey(v)) << 32) | (unsigned)i;
      atomicMin(&seg[bg], pk);
      // sorted bag_idx: segment starts need no atomics
      bool isStart = (i == 0) || (bag[i - 1] != bg);
      if (isStart) {
        sstart[bg] = i;
        multi[bg]  = (i + 1 < N && bag[i + 1] == bg) ? 1 : 0;
      }
    }
  }
}

// One wave per bag: decode packed argmax, apply the reference's sel quirk,
// write M and do float4 row gathers for bag_instances and z_ins.
__global__ void gather_kernel(const float* __restrict__ z,
                              const float* __restrict__ inst,
                              const unsigned long long* __restrict__ seg,
                              const int* __restrict__ sstart,
                              const int* __restrict__ multi,
                              float* __restrict__ outM,
                              float* __restrict__ outMI,
                              float* __restrict__ outMZ,
                              int B, int D, int F) {
  int tid  = blockIdx.x * blockDim.x + threadIdx.x;
  int g    = tid >> 5;
  int lane = threadIdx.x & 31;
  if (g >= B) return;

  unsigned long long p = seg[g];
  unsigned gidx = (unsigned)(p & 0xffffffffull);
  float vmax = fkeyinv(~(unsigned)(p >> 32));
  int sel = multi[g] ? ((int)gidx - sstart[g]) : (int)gidx;
  if (lane == 0) outM[g] = vmax;

  const float4* s1 = (const float4*)(inst + (size_t)sel * F);
  float4*       d1 = (float4*)(outMI + (size_t)g * F);
  for (int j = lane; j < (F >> 2); j += 32) d1[j] = s1[j];

  const float4* s2 = (const float4*)(z + (size_t)sel * D);
  float4*       d2 = (float4*)(outMZ + (size_t)g * D);
  for (int j = lane; j < (D >> 2); j += 32) d2[j] = s2[j];
}

extern "C" void kernel_launch(void* const* d_in, const int* in_sizes, int n_in,
                              void* d_out, int out_size, void* d_ws, size_t ws_size,
                              hipStream_t stream) {
  const float* z    = (const float*)d_in[0];
  const int*   bag  = (const int*)d_in[1];
  const float* inst = (const float*)d_in[2];
  const float* W    = (const float*)d_in[3];
  const float* bptr = (const float*)d_in[4];

  int N = in_sizes[1];                 // 500000
  int D = in_sizes[3];                 // 128
  int F = in_sizes[2] / N;             // 256
  int Bn = (int)(((long long)out_size - N) / (1LL + F + D));  // 50000

  float* outM   = (float*)d_out;                    // (B,1)
  float* outMI  = outM + Bn;                        // (B,F)
  float* outMZ  = outMI + (size_t)Bn * F;           // (B,D)
  float* outLoc = outMZ + (size_t)Bn * D;           // (N,1)

  unsigned long long* seg = (unsigned long long*)d_ws;   // B x u64
  int* sstart = (int*)(seg + Bn);                        // B x i32
  int* multi  = sstart + Bn;                             // B x i32

  seg_init_kernel<<<(Bn + 255) / 256, 256, 0, stream>>>(seg, Bn);

  int waves   = (N + 15) / 16;
  long long t = (long long)waves * 32;
  gemv_seg_kernel<<<(int)((t + 255) / 256), 256, 0, stream>>>(
      z, bag, W, bptr, outLoc, seg, sstart, multi, N, D);

  long long gt = (long long)Bn * 32;
  gather_kernel<<<(int)((gt + 255) / 256), 256, 0, stream>>>(
      z, inst, seg, sstart, multi, outM, outMI, outMZ, Bn, D, F);
}